// PositionalCombinatorOp_27144193310727
// MI455X (gfx1250) — compile-verified
//
#include <hip/hip_runtime.h>
#include <stdint.h>

// Problem shape (fixed by the reference).
constexpr int kB  = 8;
constexpr int kN  = 256;
constexpr int kMO = 64;
constexpr int kD  = 128;

// One wave32 copies one row: 32 lanes x float4 = 512 bytes = kD floats.
// Block = 256 threads = 8 waves = 8 rows. Grid = B*N*MO/8 blocks.
__global__ __launch_bounds__(256)
void PositionalCombinator_rows(const float* __restrict__ left_buf,
                               const float* __restrict__ left_count,
                               const float* __restrict__ right_buf,
                               const float* __restrict__ right_count,
                               const int*   __restrict__ subs,
                               float* __restrict__ out_buf,
                               float* __restrict__ out_cnt)
{
    // 8 waves * 512B private row slots for the async global->LDS->global copy.
    __shared__ __align__(16) float smem[8 * kD];

    const int lane = threadIdx.x & 31;
    const int wave = threadIdx.x >> 5;
    const int row  = blockIdx.x * 8 + wave;      // 0 .. B*N*MO-1 (grid covers exactly)
    const int bn   = row >> 6;                   // MO == 64
    const int pos  = row & 63;

    const float lc = left_count[bn];
    const float rc = right_count[bn];
    const int   s  = subs[bn];

    // new_count = min(first + second, MO) — symmetric in swap, write once per (b,n).
    if (pos == 0 && lane == 0) {
        out_cnt[bn] = fminf(lc + rc, (float)kMO);
    }

    const float fcf = s ? rc : lc;               // first_count
    const float scf = s ? lc : rc;               // second_count
    int fc = (int)rintf(fcf); if (fc < 0) fc = 0;   // round-half-even == jnp.round
    int sc = (int)rintf(scf); if (sc < 0) sc = 0;

    const float* first_base  = s ? right_buf : left_buf;
    const float* second_base = s ? left_buf  : right_buf;

    float* dst = out_buf + (size_t)row * kD;     // row == bn*MO + pos

    const float* src = nullptr;                  // wave-uniform selection
    if (pos < fc) {
        src = first_base + ((size_t)bn * kMO + pos) * kD;
    } else if (pos < fc + sc) {
        // pos-fc in [0, MO-1] here, so the reference's clamp is identity.
        src = second_base + ((size_t)bn * kMO + (pos - fc)) * kD;
    }

    if (src) {
        // gfx1250 async DMA row copy: global -> LDS -> global, 128b per lane.
        uint32_t lds_off = (uint32_t)(uintptr_t)(void*)&smem[wave * kD]
                         + (uint32_t)(lane * 16);
        uint64_t gsrc = (uint64_t)(uintptr_t)(src + lane * 4);
        uint64_t gdst = (uint64_t)(uintptr_t)(dst + lane * 4);

        asm volatile("global_load_async_to_lds_b128 %0, %1, off"
                     :: "v"(lds_off), "v"(gsrc) : "memory");
        asm volatile("s_wait_asynccnt 0" ::: "memory");   // data resident in LDS
        asm volatile("global_store_async_from_lds_b128 %0, %1, off"
                     :: "v"(gdst), "v"(lds_off) : "memory");
        // S_ENDPGM performs an implicit wait-idle, covering the outstanding
        // async store; LDS slot is private to this wave, so no barrier needed.
    } else {
        // Zero region: write-only, never touch the input buffers.
        float4 z = make_float4(0.f, 0.f, 0.f, 0.f);
        *reinterpret_cast<float4*>(dst + lane * 4) = z;
    }
}

extern "C" void kernel_launch(void* const* d_in, const int* in_sizes, int n_in,
                              void* d_out, int out_size, void* d_ws, size_t ws_size,
                              hipStream_t stream)
{
    const float* left_buf    = (const float*)d_in[0];
    const float* left_count  = (const float*)d_in[1];
    const float* right_buf   = (const float*)d_in[2];
    const float* right_count = (const float*)d_in[3];
    const int*   subs        = (const int*)d_in[4];

    float* out_buf = (float*)d_out;                              // (B,N,MO,D)
    float* out_cnt = out_buf + (size_t)kB * kN * kMO * kD;       // (B,N)

    const int total_rows = kB * kN * kMO;        // 131072
    const int blocks     = total_rows / 8;       // 16384, exact

    PositionalCombinator_rows<<<blocks, 256, 0, stream>>>(
        left_buf, left_count, right_buf, right_count, subs, out_buf, out_cnt);
}